// TSProjector_47596827574387
// MI455X (gfx1250) — compile-verified
//
#include <hip/hip_runtime.h>
#include <stdint.h>

typedef __attribute__((ext_vector_type(2))) float v2f;
typedef __attribute__((ext_vector_type(8))) float v8f;
typedef __attribute__((ext_vector_type(4))) unsigned int u32x4;
typedef __attribute__((ext_vector_type(4))) int i32x4;
typedef __attribute__((ext_vector_type(8))) int i32x8;

#define B_    64
#define K_    512
#define P_    64
#define T_    16384
#define SEG_  3
#define D_    1024
#define STEP_ 32

// LDS A-strip pitch: 68 floats. 68 mod 64banks = 4 -> lanes 0..15 hit banks
// 4*l and 4*l+1 for ds_load_b64 -> conflict-free. The TDM pad feature
// (interval=64 dwords, amount=4 dwords) reproduces this pitch in hardware.
#define APITCH 68

__global__ __launch_bounds__(256) void tsproj_wmma_kernel(
    const float* __restrict__ patches,
    const float* __restrict__ segments,
    const float* __restrict__ W_lin,
    const float* __restrict__ b_lin,
    const float* __restrict__ W_seg,
    const float* __restrict__ b_seg,
    const float* __restrict__ pos,
    float* __restrict__ out)
{
    __shared__ float ldsA[16 * APITCH];     // 16 x 64 patches strip (padded)
    __shared__ float ldsSeg[16 * SEG_ + 4]; // 16 x 3 pooled segment means

    const int tid  = threadIdx.x;          // 0..255
    const int m0   = blockIdx.x * 16;      // global row tile (b*K + k)
    const int nBlk = blockIdx.y * 512;     // N half
    const int b    = m0 / K_;
    const int k0   = m0 % K_;              // 16-aligned, whole tile same b

    // ---- TDM async copy: 16x64 f32 tile, padded to 68-float rows in LDS ----
    if (tid < 32) {  // wave 0 issues; EXEC ignored by TDM, TENSORcnt tracks it
        const unsigned ldsAddr = (unsigned)(uintptr_t)&ldsA[0]; // low 32b = LDS offset
        const uint64_t ga = (uint64_t)(uintptr_t)patches + (uint64_t)m0 * P_ * 4u;
        // D# group 0: count=1 | lds_addr | global_addr(57b) | type=2
        u32x4 g0 = { 1u,
                     ldsAddr,
                     (unsigned)ga,
                     ((unsigned)(ga >> 32) & 0x01FFFFFFu) | (2u << 30) };
        // D# group 1: data_size=4B, pad every 64 dwords by 4 dwords,
        // tensor_dim0=64, tensor_dim1=16, tile=64x16, dim0_stride=64
        const unsigned w0 = (2u << 16)   // data_size = 4 bytes
                          | (1u << 20)   // pad_enable
                          | (5u << 22)   // pad_interval: 2^(5+1)=64 dwords
                          | (3u << 25);  // pad_amount: 3+1 = 4 dwords
        i32x8 g1 = { (int)w0,
                     (int)(64u << 16),   // tensor_dim0[15:0] = 64
                     (int)(16u << 16),   // tensor_dim1[15:0] = 16
                     (int)(64u << 16),   // tile_dim0 = 64
                     16,                 // tile_dim1 = 16, tile_dim2 = 0
                     64,                 // tensor_dim0_stride = 64
                     0, 0 };             // tensor_dim1_stride = 0
        i32x4 gz = { 0, 0, 0, 0 };
#if __clang_major__ >= 23
        i32x8 gz8 = { 0, 0, 0, 0, 0, 0, 0, 0 };
        __builtin_amdgcn_tensor_load_to_lds(g0, g1, gz, gz, gz8, 0);
#else
        __builtin_amdgcn_tensor_load_to_lds(g0, g1, gz, gz, 0);
#endif
    }

    // ---- pooled seg_t (overlaps the TDM copy): (r,s) sums 32 strided f32 ----
    if (tid < 16 * SEG_) {
        const int r = tid / SEG_;
        const int s = tid % SEG_;
        const float* base =
            segments + ((size_t)b * T_ + (size_t)(k0 + r) * STEP_) * SEG_ + s;
        float acc = 0.0f;
#pragma unroll
        for (int j = 0; j < STEP_; ++j) acc += base[j * SEG_];
        ldsSeg[r * SEG_ + s] = acc * (1.0f / (float)STEP_);
    }

    if (tid < 32) __builtin_amdgcn_s_wait_tensorcnt(0);  // wave 0: DMA done
    __syncthreads();                                     // publish LDS to all

    const int lane   = tid & 31;
    const int wave   = tid >> 5;
    const int n_base = nBlk + wave * 64;   // this wave's 64 columns
    const int ln     = lane & 15;          // N / M-low index
    const int lh     = lane >> 4;          // lane half (K-group / M-high)

    v8f acc[4] = {v8f{}, v8f{}, v8f{}, v8f{}};

    // ---- GEMM: 16 K-steps of 4, 4 N-tiles each -> 64 WMMAs ----
#pragma unroll
    for (int kk = 0; kk < 16; ++kk) {
        const int kbase = kk * 4 + lh * 2;
        // A frag: lanes 0-15 hold K=kk*4+{0,1}, lanes 16-31 hold K=kk*4+{2,3}
        const v2f afrag = *(const v2f*)&ldsA[ln * APITCH + kbase];
        const float* wb = W_lin + kbase;   // column offset into W_lin rows
        const v2f b0 = *(const v2f*)&wb[(size_t)(n_base +  0 + ln) * P_];
        const v2f b1 = *(const v2f*)&wb[(size_t)(n_base + 16 + ln) * P_];
        const v2f b2 = *(const v2f*)&wb[(size_t)(n_base + 32 + ln) * P_];
        const v2f b3 = *(const v2f*)&wb[(size_t)(n_base + 48 + ln) * P_];
        acc[0] = __builtin_amdgcn_wmma_f32_16x16x4_f32(
            false, afrag, false, b0, (short)0, acc[0], false, false);
        acc[1] = __builtin_amdgcn_wmma_f32_16x16x4_f32(
            false, afrag, false, b1, (short)0, acc[1], false, false);
        acc[2] = __builtin_amdgcn_wmma_f32_16x16x4_f32(
            false, afrag, false, b2, (short)0, acc[2], false, false);
        acc[3] = __builtin_amdgcn_wmma_f32_16x16x4_f32(
            false, afrag, false, b3, (short)0, acc[3], false, false);
    }

    // ---- epilogue: seg_t @ W_seg^T + biases + pos, nontemporal store ----
#pragma unroll
    for (int t = 0; t < 4; ++t) {
        const int d = n_base + t * 16 + ln;
        const float ws0  = W_seg[d * SEG_ + 0];
        const float ws1  = W_seg[d * SEG_ + 1];
        const float ws2  = W_seg[d * SEG_ + 2];
        const float bias = b_lin[d] + b_seg[d];
#pragma unroll
        for (int i = 0; i < 8; ++i) {
            const int ml = i + lh * 8;  // local row: VGPR i, lane half lh
            const float s0 = ldsSeg[ml * SEG_ + 0];
            const float s1 = ldsSeg[ml * SEG_ + 1];
            const float s2 = ldsSeg[ml * SEG_ + 2];
            const float pv = pos[(size_t)(k0 + ml) * D_ + d];
            const float v  = acc[t][i] + bias + pv
                           + s0 * ws0 + s1 * ws1 + s2 * ws2;
            __builtin_nontemporal_store(v, &out[(size_t)(m0 + ml) * D_ + d]);
        }
    }
}

extern "C" void kernel_launch(void* const* d_in, const int* in_sizes, int n_in,
                              void* d_out, int out_size, void* d_ws, size_t ws_size,
                              hipStream_t stream) {
    const float* patches  = (const float*)d_in[0];
    const float* segments = (const float*)d_in[1];
    const float* W_lin    = (const float*)d_in[2];
    const float* b_lin    = (const float*)d_in[3];
    const float* W_seg    = (const float*)d_in[4];
    const float* b_seg    = (const float*)d_in[5];
    const float* pos      = (const float*)d_in[6];
    float* out = (float*)d_out;

    dim3 grid((B_ * K_) / 16, D_ / 512, 1);  // (2048, 2)
    tsproj_wmma_kernel<<<grid, 256, 0, stream>>>(
        patches, segments, W_lin, b_lin, W_seg, b_seg, pos, out);
}